// GraphConv_42331197669585
// MI455X (gfx1250) — compile-verified
//
#include <hip/hip_runtime.h>

typedef __attribute__((ext_vector_type(2))) float v2f;
typedef __attribute__((ext_vector_type(8))) float v8f;

#define D 128
#define TILE_M 16

// ---------------------------------------------------------------------------
// Kernel 1: zero the scatter accumulator and degree counter in workspace.
// ---------------------------------------------------------------------------
__global__ void init_ws_kernel(float* __restrict__ acc, float* __restrict__ deg,
                               size_t n_acc, size_t total) {
    size_t stride = (size_t)gridDim.x * blockDim.x;
    for (size_t i = (size_t)blockIdx.x * blockDim.x + threadIdx.x; i < total; i += stride) {
        if (i < n_acc) acc[i] = 0.0f;
        else           deg[i - n_acc] = 0.0f;
    }
}

// ---------------------------------------------------------------------------
// Kernel 2: dual GEMM via FP32 WMMA. One wave (32 lanes) per 16-row tile.
//   h   = x @ W            -> workspace (feeds the edge scatter)
//   out = x @ R + bias     -> d_out (aggregated term added later)
//
// A-matrix 16x4 f32 layout (per ISA 7.12.2):
//   lane L (0..15):  a.x = A[M=L][K=4k+0], a.y = A[M=L][K=4k+1]
//   lane L+16:       a.x = A[M=L][K=4k+2], a.y = A[M=L][K=4k+3]
// B-matrix 4x16 mirrors this (K split across lane halves, N = lane&15).
// C/D 16x16: element v of v8f = row (v + 8*(lane>=16)), col (lane&15).
// ---------------------------------------------------------------------------
__global__ __launch_bounds__(32) void gemm_dual_kernel(
    const float* __restrict__ x, const float* __restrict__ W,
    const float* __restrict__ R, const float* __restrict__ bias,
    float* __restrict__ h, float* __restrict__ out) {
    const int row0 = blockIdx.x * TILE_M;
    const int lane = threadIdx.x;
    const int lo = lane & 15;
    const int hi = lane >> 4;

    // Preload all 32 A fragments for this 16-row tile (K = 0..127, 4 per step).
    v2f a[32];
    const float* xrow = x + (size_t)(row0 + lo) * D + 2 * hi;
    #pragma unroll
    for (int k = 0; k < 32; ++k) {
        a[k] = *(const v2f*)(xrow + 4 * k);   // 8B-aligned: index is even
    }

    #pragma unroll 1
    for (int t = 0; t < 8; ++t) {           // 8 output column tiles of 16
        const int n0 = t * 16;
        v8f cw = {};
        v8f cr = {};
        const float* wp = W + (size_t)(2 * hi) * D + n0 + lo;
        const float* rp = R + (size_t)(2 * hi) * D + n0 + lo;
        #pragma unroll
        for (int k = 0; k < 32; ++k) {
            v2f bw, br;
            bw.x = wp[(size_t)(4 * k) * D];
            bw.y = wp[(size_t)(4 * k + 1) * D];
            br.x = rp[(size_t)(4 * k) * D];
            br.y = rp[(size_t)(4 * k + 1) * D];
            cw = __builtin_amdgcn_wmma_f32_16x16x4_f32(
                false, a[k], false, bw, (short)0, cw, false, false);
            cr = __builtin_amdgcn_wmma_f32_16x16x4_f32(
                false, a[k], false, br, (short)0, cr, false, false);
        }
        const float bv = bias[n0 + lo];
        const size_t base = (size_t)(row0 + 8 * hi) * D + n0 + lo;
        #pragma unroll
        for (int v = 0; v < 8; ++v) {
            h[base + (size_t)v * D]   = cw[v];
            out[base + (size_t)v * D] = cr[v] + bv;
        }
    }
}

// ---------------------------------------------------------------------------
// Kernel 3: edge scatter. One wave32 per edge:
//   acc[row][:] += h[col][:]  (4 coalesced f32 atomics per lane)
//   deg[row]    += 1          (lane 0)
// ---------------------------------------------------------------------------
__global__ __launch_bounds__(256) void edge_scatter_kernel(
    const long long* __restrict__ ei, const float* __restrict__ h,
    float* __restrict__ acc, float* __restrict__ deg, int n_edges) {
    const int lane = threadIdx.x & 31;
    const int wid  = threadIdx.x >> 5;
    const int e = blockIdx.x * (blockDim.x >> 5) + wid;
    if (e >= n_edges) return;

    const int row = (int)ei[e];                       // destination
    const int col = (int)ei[(size_t)n_edges + e];     // source (message)
    const float* hs = h + (size_t)col * D;
    float* ad = acc + (size_t)row * D;
    #pragma unroll
    for (int k = 0; k < 4; ++k) {
        atomicAdd(ad + lane + 32 * k, hs[lane + 32 * k]);
    }
    if (lane == 0) atomicAdd(deg + row, 1.0f);
}

// ---------------------------------------------------------------------------
// Kernel 4: out += acc / max(deg, 1)
// ---------------------------------------------------------------------------
__global__ void finalize_kernel(const float* __restrict__ acc,
                                const float* __restrict__ deg,
                                float* __restrict__ out, size_t n_acc) {
    size_t stride = (size_t)gridDim.x * blockDim.x;
    for (size_t i = (size_t)blockIdx.x * blockDim.x + threadIdx.x; i < n_acc; i += stride) {
        out[i] += acc[i] / fmaxf(deg[i >> 7], 1.0f);
    }
}

extern "C" void kernel_launch(void* const* d_in, const int* in_sizes, int n_in,
                              void* d_out, int out_size, void* d_ws, size_t ws_size,
                              hipStream_t stream) {
    const float*     x    = (const float*)d_in[0];
    const long long* ei   = (const long long*)d_in[1];   // int64 [2, E]
    const float*     W    = (const float*)d_in[2];
    const float*     R    = (const float*)d_in[3];
    const float*     bias = (const float*)d_in[4];
    float* out = (float*)d_out;

    const int n_nodes = in_sizes[0] / D;
    const int n_edges = in_sizes[1] / 2;

    char* ws = (char*)d_ws;
    float* h   = (float*)ws;
    float* acc = (float*)(ws + (size_t)n_nodes * D * sizeof(float));
    float* deg = (float*)(ws + 2 * (size_t)n_nodes * D * sizeof(float));

    const size_t n_acc = (size_t)n_nodes * D;

    init_ws_kernel<<<2048, 256, 0, stream>>>(acc, deg, n_acc, n_acc + n_nodes);

    gemm_dual_kernel<<<n_nodes / TILE_M, 32, 0, stream>>>(x, W, R, bias, h, out);

    const int waves_per_block = 8;  // 256 threads = 8 wave32
    const int eb = (n_edges + waves_per_block - 1) / waves_per_block;
    edge_scatter_kernel<<<eb, 256, 0, stream>>>(ei, h, acc, deg, n_edges);

    finalize_kernel<<<2048, 256, 0, stream>>>(acc, deg, out, n_acc);
}